// RelGraphConvLayer_74302934221479
// MI455X (gfx1250) — compile-verified
//
#include <hip/hip_runtime.h>
#include <hip/hip_bf16.h>

// ---------------------------------------------------------------------------
// R-GCN basis-decomposition layer for gfx1250 (MI455X), fp32 WMMA path.
//   out = relu( scatter_add(norm_e * x[src_e] @ W_{etype_e}, dst)
//               + h_bias + x @ loop_weight )
//   W_r = sum_b w_comp[r,b] * V_b      (precomputed once: 64x128x128 = 4MB)
//
// Edges are bucketed by relation; each block processes a 64-edge tile of one
// relation: 8 waves x (4 stacked 16x16 accumulators) so every B fragment of
// W_r is reused 4x in registers. All message FLOPs flow through
// v_wmma_f32_16x16x4_f32 (fp32 end-to-end to match the fp32 reference).
// ---------------------------------------------------------------------------

typedef __attribute__((ext_vector_type(2))) float v2f;
typedef __attribute__((ext_vector_type(8))) float v8f;

#define IN_FEAT   128
#define OUT_FEAT  128
#define NUM_RELS  64
#define NUM_BASES 8
#define TILE_E    64          // edges (or nodes) per block
#define LDS_STRIDE 132        // 132 % 64 == 4 -> conflict-free A-fragment reads

// ------------------------- small utility kernels ---------------------------

__global__ void zero_meta_kernel(int* meta) {
    int i = threadIdx.x;
    if (i < NUM_RELS) meta[i] = 0;    // counts
    if (i == NUM_RELS) meta[193] = 0; // totalTiles
}

__global__ void hist_kernel(const int* __restrict__ etypes, int nE,
                            int* __restrict__ counts) {
    int e = blockIdx.x * 256 + threadIdx.x;
    if (e < nE) atomicAdd(&counts[etypes[e]], 1);
}

__global__ void scan_kernel(int* meta) {
    if (threadIdx.x == 0) {
        int* counts  = meta;
        int* offsets = meta + 64;   // 65 entries
        int* cursor  = meta + 129;  // 64 entries
        int s = 0;
        for (int r = 0; r < NUM_RELS; ++r) {
            offsets[r] = s;
            cursor[r]  = s;
            s += counts[r];
        }
        offsets[NUM_RELS] = s;
    }
}

__global__ void scatter_kernel(const int* __restrict__ etypes, int nE,
                               int* __restrict__ cursor,
                               int* __restrict__ perm) {
    int e = blockIdx.x * 256 + threadIdx.x;
    if (e < nE) {
        int p = atomicAdd(&cursor[etypes[e]], 1);
        perm[p] = e;
    }
}

__global__ void tiles_kernel(int* __restrict__ meta,
                             int* __restrict__ tileRel,
                             int* __restrict__ tileStart,
                             int* __restrict__ tileCnt) {
    __shared__ int ntoff[NUM_RELS + 1];
    int* counts  = meta;
    int* offsets = meta + 64;
    if (threadIdx.x == 0) {
        int s = 0;
        for (int r = 0; r < NUM_RELS; ++r) {
            ntoff[r] = s;
            s += (counts[r] + TILE_E - 1) / TILE_E;
        }
        ntoff[NUM_RELS] = s;
        meta[193] = s; // totalTiles
    }
    __syncthreads();
    int r = threadIdx.x;
    if (r < NUM_RELS) {
        int cnt  = counts[r];
        int nt   = (cnt + TILE_E - 1) / TILE_E;
        int base = ntoff[r];
        int off  = offsets[r];
        for (int t = 0; t < nt; ++t) {
            int idx = base + t;
            tileRel[idx]   = r;
            tileStart[idx] = off + TILE_E * t;
            int rem = cnt - TILE_E * t;
            tileCnt[idx]   = rem < TILE_E ? rem : TILE_E;
        }
    }
}

// W_rel[r][i][j] = sum_b w_comp[r][b] * weight[b][i][j]
__global__ void wrel_kernel(const float* __restrict__ weight,
                            const float* __restrict__ w_comp,
                            float* __restrict__ Wrel) {
    int idx = blockIdx.x * 256 + threadIdx.x; // < 64*128*128
    int r  = idx >> 14;
    int ij = idx & 16383;
    float s = 0.f;
#pragma unroll
    for (int b = 0; b < NUM_BASES; ++b)
        s += w_comp[r * NUM_BASES + b] * weight[b * 16384 + ij];
    Wrel[idx] = s;
}

__global__ void relu_kernel(float* __restrict__ out, int n) {
    int i = blockIdx.x * 256 + threadIdx.x;
    if (i < n) out[i] = fmaxf(out[i], 0.f);
}

// ------------------------- WMMA GEMM kernels -------------------------------
// Block = 256 threads = 8 wave32; wave w owns output columns [16w, 16w+16)
// and 4 stacked 16-row sub-tiles (64 rows total), so each B fragment is
// reused by 4 WMMAs. A (64 x 128 K) staged in LDS with stride 132
// (132 % 64 == 4 -> the 16 lanes of an A-fragment read hit distinct banks).
//
// v_wmma_f32_16x16x4_f32 fragment mapping (ISA 7.12.2):
//   A: lane L -> M = L&15, K = kstep + 2*(L>>4) + {vgpr0,vgpr1}
//   B: lane L -> N = L&15, K = kstep + 2*(L>>4) + {vgpr0,vgpr1}
//   D: vgpr j, lane L -> row 8*(L>>4)+j, col L&15

// Self-loop: out[node] = x[node] @ loop_weight + h_bias   (dense, non-atomic)
__global__ __launch_bounds__(256) void selfloop_gemm_kernel(
        const float* __restrict__ x, const float* __restrict__ Wl,
        const float* __restrict__ bias, float* __restrict__ out, int nNodes) {
    __shared__ float Zs[TILE_E][LDS_STRIDE];
    const int base = blockIdx.x * TILE_E;

    // Stage 64 node rows into LDS, float4-vectorized (rows are 512B-aligned).
    for (int i = threadIdx.x; i < TILE_E * (IN_FEAT / 4); i += 256) {
        int row = i >> 5, c4 = (i & 31) << 2;
        int node = base + row;
        float4 v = make_float4(0.f, 0.f, 0.f, 0.f);
        if (node < nNodes)
            v = *(const float4*)&x[(size_t)node * IN_FEAT + c4];
        *(float4*)&Zs[row][c4] = v;
    }
    __syncthreads();

    const int lane  = threadIdx.x & 31;
    const int wave  = threadIdx.x >> 5;
    const int m     = lane & 15;            // A row-within-subtile / B-D col
    const int khalf = (lane >> 4) << 1;     // K offset for hi half-wave
    const int col   = wave * 16 + m;        // global output column

    v8f acc0 = {}, acc1 = {}, acc2 = {}, acc3 = {};
#pragma unroll
    for (int k = 0; k < IN_FEAT; k += 4) {
        const int kk = k + khalf;
        v2f b;
        b.x = Wl[(size_t)kk * OUT_FEAT + col];
        b.y = Wl[(size_t)(kk + 1) * OUT_FEAT + col];
        v2f a0, a1, a2, a3;
        a0.x = Zs[m     ][kk]; a0.y = Zs[m     ][kk + 1];
        a1.x = Zs[m + 16][kk]; a1.y = Zs[m + 16][kk + 1];
        a2.x = Zs[m + 32][kk]; a2.y = Zs[m + 32][kk + 1];
        a3.x = Zs[m + 48][kk]; a3.y = Zs[m + 48][kk + 1];
        acc0 = __builtin_amdgcn_wmma_f32_16x16x4_f32(false, a0, false, b, (short)0, acc0, false, false);
        acc1 = __builtin_amdgcn_wmma_f32_16x16x4_f32(false, a1, false, b, (short)0, acc1, false, false);
        acc2 = __builtin_amdgcn_wmma_f32_16x16x4_f32(false, a2, false, b, (short)0, acc2, false, false);
        acc3 = __builtin_amdgcn_wmma_f32_16x16x4_f32(false, a3, false, b, (short)0, acc3, false, false);
    }

    const float bcol  = bias[col];
    const int   rbase = (lane >> 4) * 8;
#pragma unroll
    for (int j = 0; j < 8; ++j) {
        int r0 = base + rbase + j;
        if (r0      < nNodes) out[(size_t)(r0     ) * OUT_FEAT + col] = acc0[j] + bcol;
        if (r0 + 16 < nNodes) out[(size_t)(r0 + 16) * OUT_FEAT + col] = acc1[j] + bcol;
        if (r0 + 32 < nNodes) out[(size_t)(r0 + 32) * OUT_FEAT + col] = acc2[j] + bcol;
        if (r0 + 48 < nNodes) out[(size_t)(r0 + 48) * OUT_FEAT + col] = acc3[j] + bcol;
    }
}

// Edge messages: one block per 64-edge tile of a single relation bucket.
// msg = (x[src] @ W_rel[r]) * norm; atomicAdd into out[dst].
__global__ __launch_bounds__(256) void edge_msg_kernel(
        const float* __restrict__ x, const float* __restrict__ Wrel,
        const float* __restrict__ norm,
        const int* __restrict__ src, const int* __restrict__ dst,
        const int* __restrict__ perm,
        const int* __restrict__ tileRel, const int* __restrict__ tileStart,
        const int* __restrict__ tileCnt, const int* __restrict__ totalTiles,
        float* __restrict__ out) {
    const int t = blockIdx.x;
    if (t >= *totalTiles) return;   // uniform per block: EXEC stays full

    __shared__ float Zs[TILE_E][LDS_STRIDE];
    __shared__ float norm_s[TILE_E];
    __shared__ int   dst_s[TILE_E];
    __shared__ int   node_s[TILE_E];   // src node per row, -1 for padding

    const int rel   = tileRel[t];
    const int start = tileStart[t];
    const int cnt   = tileCnt[t];
    const float* __restrict__ W = Wrel + (size_t)rel * IN_FEAT * OUT_FEAT;

    if (threadIdx.x < TILE_E) {
        int row = threadIdx.x;
        if (row < cnt) {
            int e = perm[start + row];
            node_s[row] = src[e];
            norm_s[row] = norm[e];
            dst_s[row]  = dst[e];
        } else {
            node_s[row] = -1;
            norm_s[row] = 0.f;
            dst_s[row]  = -1;
        }
    }
    // Warm the relation's weight block toward the caches (global_prefetch_b8).
    __builtin_prefetch(W + (threadIdx.x << 6), 0, 3);
    __syncthreads();

    // Gather 64 source rows into LDS, float4-vectorized.
    for (int i = threadIdx.x; i < TILE_E * (IN_FEAT / 4); i += 256) {
        int row = i >> 5, c4 = (i & 31) << 2;
        int node = node_s[row];
        float4 v = make_float4(0.f, 0.f, 0.f, 0.f);
        if (node >= 0)
            v = *(const float4*)&x[(size_t)node * IN_FEAT + c4];
        *(float4*)&Zs[row][c4] = v;
    }
    __syncthreads();

    const int lane  = threadIdx.x & 31;
    const int wave  = threadIdx.x >> 5;
    const int m     = lane & 15;
    const int khalf = (lane >> 4) << 1;
    const int col   = wave * 16 + m;

    v8f acc0 = {}, acc1 = {}, acc2 = {}, acc3 = {};
#pragma unroll
    for (int k = 0; k < IN_FEAT; k += 4) {
        const int kk = k + khalf;
        v2f b;
        b.x = W[(size_t)kk * OUT_FEAT + col];
        b.y = W[(size_t)(kk + 1) * OUT_FEAT + col];
        v2f a0, a1, a2, a3;
        a0.x = Zs[m     ][kk]; a0.y = Zs[m     ][kk + 1];
        a1.x = Zs[m + 16][kk]; a1.y = Zs[m + 16][kk + 1];
        a2.x = Zs[m + 32][kk]; a2.y = Zs[m + 32][kk + 1];
        a3.x = Zs[m + 48][kk]; a3.y = Zs[m + 48][kk + 1];
        acc0 = __builtin_amdgcn_wmma_f32_16x16x4_f32(false, a0, false, b, (short)0, acc0, false, false);
        acc1 = __builtin_amdgcn_wmma_f32_16x16x4_f32(false, a1, false, b, (short)0, acc1, false, false);
        acc2 = __builtin_amdgcn_wmma_f32_16x16x4_f32(false, a2, false, b, (short)0, acc2, false, false);
        acc3 = __builtin_amdgcn_wmma_f32_16x16x4_f32(false, a3, false, b, (short)0, acc3, false, false);
    }

    const int rbase = (lane >> 4) * 8;
#pragma unroll
    for (int j = 0; j < 8; ++j) {
        int r0 = rbase + j;
        if (r0 < cnt)
            atomicAdd(&out[(size_t)dst_s[r0] * OUT_FEAT + col], acc0[j] * norm_s[r0]);
        int r1 = r0 + 16;
        if (r1 < cnt)
            atomicAdd(&out[(size_t)dst_s[r1] * OUT_FEAT + col], acc1[j] * norm_s[r1]);
        int r2 = r0 + 32;
        if (r2 < cnt)
            atomicAdd(&out[(size_t)dst_s[r2] * OUT_FEAT + col], acc2[j] * norm_s[r2]);
        int r3 = r0 + 48;
        if (r3 < cnt)
            atomicAdd(&out[(size_t)dst_s[r3] * OUT_FEAT + col], acc3[j] * norm_s[r3]);
    }
}

// ------------------------------ launcher -----------------------------------

extern "C" void kernel_launch(void* const* d_in, const int* in_sizes, int n_in,
                              void* d_out, int out_size, void* d_ws, size_t ws_size,
                              hipStream_t stream) {
    // Input order from setup_inputs():
    const float* x        = (const float*)d_in[0];  // [N,128]
    const float* norm     = (const float*)d_in[1];  // [E,1]
    const float* weight   = (const float*)d_in[2];  // [8,128,128]
    const float* w_comp   = (const float*)d_in[3];  // [64,8]
    const float* h_bias   = (const float*)d_in[4];  // [128]
    const float* loop_w   = (const float*)d_in[5];  // [128,128]
    const int*   src      = (const int*)d_in[6];    // [E]
    const int*   dst      = (const int*)d_in[7];    // [E]
    const int*   etypes   = (const int*)d_in[8];    // [E]

    float* out = (float*)d_out;

    const int nNodes = in_sizes[0] / IN_FEAT;
    const int nE     = in_sizes[6];
    const int maxTiles = nE / TILE_E + NUM_RELS + 1;

    // ---- workspace carving (offsets 256B-aligned up to tileRel) ----
    char* ws = (char*)d_ws;
    size_t off = 0;
    float* Wrel = (float*)(ws + off);                       // 64*128*128 f32
    off += (size_t)NUM_RELS * IN_FEAT * OUT_FEAT * sizeof(float);
    int* perm = (int*)(ws + off);                           // [E]
    off += ((size_t)nE * sizeof(int) + 255) & ~(size_t)255;
    int* meta = (int*)(ws + off);                           // 256 ints:
    off += 256 * sizeof(int);                               //  [0..63] counts
    int* counts  = meta;                                    //  [64..128] offsets
    int* cursor  = meta + 129;                              //  [129..192] cursor
    int* totalTiles = meta + 193;                           //  [193] totalTiles
    int* tileRel   = (int*)(ws + off); off += (size_t)maxTiles * sizeof(int);
    int* tileStart = (int*)(ws + off); off += (size_t)maxTiles * sizeof(int);
    int* tileCnt   = (int*)(ws + off); off += (size_t)maxTiles * sizeof(int);
    (void)ws_size; (void)n_in; (void)out_size;

    const int edgeBlocks = (nE + 255) / 256;

    // 1) per-relation weights  W_r = sum_b w_comp[r,b] * V_b
    wrel_kernel<<<(NUM_RELS * IN_FEAT * OUT_FEAT) / 256, 256, 0, stream>>>(
        weight, w_comp, Wrel);

    // 2) bucket edges by relation
    zero_meta_kernel<<<1, 128, 0, stream>>>(meta);
    hist_kernel<<<edgeBlocks, 256, 0, stream>>>(etypes, nE, counts);
    scan_kernel<<<1, 64, 0, stream>>>(meta);
    scatter_kernel<<<edgeBlocks, 256, 0, stream>>>(etypes, nE, cursor, perm);
    tiles_kernel<<<1, 64, 0, stream>>>(meta, tileRel, tileStart, tileCnt);

    // 3) out = x @ loop_weight + h_bias   (initializes d_out)
    selfloop_gemm_kernel<<<(nNodes + TILE_E - 1) / TILE_E, 256, 0, stream>>>(
        x, loop_w, h_bias, out, nNodes);

    // 4) scatter-add relation-bucketed WMMA messages
    edge_msg_kernel<<<maxTiles, 256, 0, stream>>>(
        x, Wrel, norm, src, dst, perm,
        tileRel, tileStart, tileCnt, totalTiles, out);

    // 5) ReLU
    const int nOut = nNodes * OUT_FEAT;
    relu_kernel<<<(nOut + 255) / 256, 256, 0, stream>>>(out, nOut);
}